// PoolG_57329223467460
// MI455X (gfx1250) — compile-verified
//
#include <hip/hip_runtime.h>

#define UNITS 256          // columns per row (fixed by the reference)
#define TILE  16           // rows staged per LDS tile (16 KB payload)
#define NBUF  3            // triple buffer: 2 tiles in flight while computing 1
#define LROW  272          // padded LDS row stride in floats -> bank-conflict-free fragments
#define BLOCK 512          // 16 waves
#define PER_THREAD 2       // (TILE * UNITS/4) / BLOCK  float4 copies per thread per tile

typedef __attribute__((ext_vector_type(2))) float v2f;
typedef __attribute__((ext_vector_type(8))) float v8f;

#if defined(__gfx1250__) && \
    __has_builtin(__builtin_amdgcn_global_load_async_to_lds_b128) && \
    __has_builtin(__builtin_amdgcn_s_wait_asynccnt)
#define USE_ASYNC 1
typedef __attribute__((ext_vector_type(4))) int v4i;
typedef __attribute__((address_space(1))) v4i* gv4_t;   // global-memory int4*
typedef __attribute__((address_space(3))) v4i* lv4_t;   // LDS int4*
#else
#define USE_ASYNC 0
#endif

// ---------------------------------------------------------------------------
// Kernel 1: inclusive prefix sum of segment sizes (sizes[i] = nc[i]*nf[i]).
// ---------------------------------------------------------------------------
__global__ void segmean_scan_kernel(const int* __restrict__ nclasses,
                                    const int* __restrict__ nfeature,
                                    int* __restrict__ cum) {
    __shared__ int sm[256];
    const int t = threadIdx.x;
    sm[t] = nclasses[t] * nfeature[t];
    __syncthreads();
#pragma unroll
    for (int off = 1; off < 256; off <<= 1) {
        int v = (t >= off) ? sm[t - off] : 0;
        __syncthreads();
        sm[t] += v;
        __syncthreads();
    }
    cum[t] = sm[t];
}

// ---------------------------------------------------------------------------
// Kernel 2: streaming column-sum of one row-slice of one segment.
//   grid = (segments, SPLIT), block = 512 = 16 waves; wave w owns cols [16w,16w+16)
//
// Main tiles: 16 contiguous rows (16 KB) async-copied global->LDS
// (global_load_async_to_lds_b128, fully coalesced, ASYNCcnt), TRIPLE-buffered:
// two tiles stay in flight (s_wait_asynccnt 4) while one is consumed.
// Accumulate: V_WMMA_F32_16X16X4_F32 with A = ones(16x4):
//   D[m][n] += sum_k B[k][n]  -> 4 rows x 16 cols per instruction, f32 exact.
// B fragment (f32 4x16): b.x lanes0-15=(K0,N=lane) lanes16-31=(K2,N=lane-16);
//                        b.y K1 / K3.  D: VGPR0 lanes0-15 = (M=0,N=lane).
// LDS row stride 272 floats => fragment banks (col, col+32): conflict-free.
// ---------------------------------------------------------------------------
__global__ __launch_bounds__(BLOCK)
void segmean_sum_kernel(const float* __restrict__ x,
                        const int* __restrict__ cum,
                        float* __restrict__ dst,
                        int do_div) {
    __shared__ float buf[NBUF][TILE * LROW];   // 3 * 17408 B = 52224 B

    const int seg   = blockIdx.x;
    const int s     = blockIdx.y;
    const int split = (int)gridDim.y;

    const int start = (seg == 0) ? 0 : cum[seg - 1];
    const int rows  = cum[seg] - start;

    const int rps = (rows + split - 1) / split;
    const int r0  = s * rps;
    int       r1  = r0 + rps;
    if (r1 > rows) r1 = rows;

    const int tid  = threadIdx.x;
    const int lane = tid & 31;
    const int wave = tid >> 5;
    const int col0 = wave * 16;
    const int half = lane >> 4;
    const int n    = lane & 15;

    const float* sbase = x + ((long)start + r0) * UNITS;   // first row of slice

    v8f c = {0.f, 0.f, 0.f, 0.f, 0.f, 0.f, 0.f, 0.f};
    v2f a = {1.f, 1.f};                                    // ones A-matrix

    const int ntiles = (r1 - r0) / TILE;

    // ---- WMMA consume of one staged tile ----
    auto compute = [&](int b) {
        const float* lb = &buf[b][col0 + n];
#pragma unroll
        for (int g = 0; g < TILE / 4; ++g) {
            v2f bb;
            bb.x = lb[(4 * g + 2 * half)     * LROW];
            bb.y = lb[(4 * g + 2 * half + 1) * LROW];
            c = __builtin_amdgcn_wmma_f32_16x16x4_f32(
                    false, a, false, bb, (short)0, c, false, false);
        }
    };

#if USE_ASYNC
    // ---- async global->LDS staging, triple buffered ----
    auto issue = [&](int T, int b) {
#pragma unroll
        for (int j = 0; j < PER_THREAD; ++j) {
            const int f   = j * BLOCK + tid;      // float4 index within tile
            const int row = f >> 6;               // 64 float4 per 1KB row
            const int c4  = f & 63;
            const float* gp = sbase + (long)(T * TILE + row) * UNITS + c4 * 4;
            float*       lp = &buf[b][row * LROW + c4 * 4];
            __builtin_amdgcn_global_load_async_to_lds_b128(
                (gv4_t)gp, (lv4_t)lp, 0, 0);
        }
    };

    if (ntiles > 0) {
        issue(0, 0);
        if (ntiles > 1) issue(1, 1);
        for (int t = 0; t < ntiles; ++t) {
            if (t + 2 < ntiles) {
                issue(t + 2, (t + 2) % NBUF);
                __builtin_amdgcn_s_wait_asynccnt(2 * PER_THREAD); // tiles t+1,t+2 in flight
            } else if (t + 1 < ntiles) {
                __builtin_amdgcn_s_wait_asynccnt(PER_THREAD);     // tile t+1 in flight
            } else {
                __builtin_amdgcn_s_wait_asynccnt(0);
            }
            __syncthreads();          // all waves' tile-t data visible
            compute(t % NBUF);
            __syncthreads();          // reads done before this buffer is reissued
        }
    }
#else
    // ---- fallback: register-staged double buffer with b128 loads ----
    if (ntiles > 0) {
        float4 st[PER_THREAD];
        auto load_stage = [&](int T) {
#pragma unroll
            for (int j = 0; j < PER_THREAD; ++j) {
                const int f   = j * BLOCK + tid;
                const int row = f >> 6;
                const int c4  = f & 63;
                st[j] = *(const float4*)(sbase + (long)(T * TILE + row) * UNITS + c4 * 4);
            }
        };
        auto store_stage = [&](int b) {
#pragma unroll
            for (int j = 0; j < PER_THREAD; ++j) {
                const int f   = j * BLOCK + tid;
                const int row = f >> 6;
                const int c4  = f & 63;
                *(float4*)&buf[b][row * LROW + c4 * 4] = st[j];
            }
        };
        load_stage(0);
        store_stage(0);
        __syncthreads();
        for (int t = 0; t < ntiles; ++t) {
            if (t + 1 < ntiles) load_stage(t + 1);   // deep MLP while computing
            compute(t & 1);
            __syncthreads();
            if (t + 1 < ntiles) store_stage((t + 1) & 1);
            __syncthreads();
        }
    }
#endif

    // ---- tail rows (< TILE): guarded global fragments ----
    {
        const float* basep = x + (long)start * UNITS + col0 + n;
        int r = r0 + ntiles * TILE;
        for (; r + 4 <= r1; r += 4) {
            v2f bb;
            bb.x = basep[(long)(r + 2 * half)     * UNITS];
            bb.y = basep[(long)(r + 2 * half + 1) * UNITS];
            c = __builtin_amdgcn_wmma_f32_16x16x4_f32(
                    false, a, false, bb, (short)0, c, false, false);
        }
        if (r < r1) {
            const int ra = r + 2 * half;
            const int rb = ra + 1;
            v2f bb;
            bb.x = (ra < r1) ? basep[(long)ra * UNITS] : 0.f;
            bb.y = (rb < r1) ? basep[(long)rb * UNITS] : 0.f;
            c = __builtin_amdgcn_wmma_f32_16x16x4_f32(
                    false, a, false, bb, (short)0, c, false, false);
        }
    }

    if (lane < 16) {
        float v = c[0];                          // M=0 row of D = column sums
        if (do_div) v = v / (float)rows;
        dst[((long)seg * split + s) * UNITS + col0 + n] = v;
    }
}

// ---------------------------------------------------------------------------
// Kernel 3: combine SPLIT deterministic partials and divide by segment size.
// ---------------------------------------------------------------------------
__global__ void segmean_reduce_kernel(const float* __restrict__ partials,
                                      const int* __restrict__ cum,
                                      float* __restrict__ out,
                                      int split) {
    const int seg = blockIdx.x;
    const int col = threadIdx.x;
    const int startv = (seg == 0) ? 0 : cum[seg - 1];
    const int rows   = cum[seg] - startv;
    float s = 0.f;
    for (int i = 0; i < split; ++i)
        s += partials[((long)seg * split + i) * UNITS + col];
    out[(long)seg * UNITS + col] = s / (float)rows;
}

// ---------------------------------------------------------------------------
extern "C" void kernel_launch(void* const* d_in, const int* in_sizes, int n_in,
                              void* d_out, int out_size, void* d_ws, size_t ws_size,
                              hipStream_t stream) {
    const float* x   = (const float*)d_in[0];
    const int*   nc  = (const int*)d_in[1];
    const int*   nf  = (const int*)d_in[2];
    float*       out = (float*)d_out;

    const int B = in_sizes[1];   // 256 segments

    int* cum = (int*)d_ws;
    const size_t head = ((size_t)B * sizeof(int) + 255) & ~(size_t)255;
    float* partials = (float*)((char*)d_ws + head);

    const size_t per_split = (size_t)B * UNITS * sizeof(float);
    int split = 0;
    if (ws_size > head) split = (int)((ws_size - head) / per_split);
    if (split > 8) split = 8;    // halves the worst-case slice vs 4, 2048 blocks

    segmean_scan_kernel<<<1, B, 0, stream>>>(nc, nf, cum);

    if (split >= 1) {
        dim3 grid(B, split);
        segmean_sum_kernel<<<grid, BLOCK, 0, stream>>>(x, cum, partials, 0);
        segmean_reduce_kernel<<<B, UNITS, 0, stream>>>(partials, cum, out, split);
    } else {
        dim3 grid(B, 1);
        segmean_sum_kernel<<<grid, BLOCK, 0, stream>>>(x, cum, out, 1);
    }
}